// Attention_18717467476618
// MI455X (gfx1250) — compile-verified
//
#include <hip/hip_runtime.h>
#include <math.h>

// MI455X (gfx1250) — memory-bound GAT attention-score kernel, v2.1.
// h traffic = 819 MB @ 23.3 TB/s => ~35us floor. fp32 WMMA for the dot
// products, ASYNC global->LDS (no VGPR staging) with a double-buffered
// per-wave pipeline so each wave overlaps HBM streaming with WMMA.

typedef __attribute__((ext_vector_type(2))) float v2f;
typedef __attribute__((ext_vector_type(8))) float v8f;
typedef int b128v __attribute__((vector_size(16)));  // builtin's pointee type

#define D_DIM 128
#define K_MB  32
#define CHUNK 16    // rows per pipeline unit (one WMMA M-tile)
#define PITCH 132   // 132 mod 64 == 4 -> conflict-free strided A-fragment reads

// ---- CDNA5 async VMEM->LDS helpers (builtin if present, else inline asm) ----
__device__ __forceinline__ void async_copy_b128(const float* g, float* l) {
#if __has_builtin(__builtin_amdgcn_global_load_async_to_lds_b128)
  __builtin_amdgcn_global_load_async_to_lds_b128(
      (__attribute__((address_space(1))) b128v*)(b128v*)(void*)const_cast<float*>(g),
      (__attribute__((address_space(3))) b128v*)(b128v*)(void*)l,
      0, 0);
#else
  unsigned loff = (unsigned)(size_t)(__attribute__((address_space(3))) void*)l;
  asm volatile("global_load_async_to_lds_b128 %0, %1, off"
               :: "v"(loff), "v"((unsigned long long)(size_t)g)
               : "memory");
#endif
}

template <int CNT>
__device__ __forceinline__ void wait_asynccnt() {
#if __has_builtin(__builtin_amdgcn_s_wait_asynccnt)
  __builtin_amdgcn_s_wait_asynccnt(CNT);
#else
  asm volatile("s_wait_asynccnt %0" :: "n"(CNT) : "memory");
#endif
  asm volatile("" ::: "memory");  // compile-time barrier: no LDS-read hoisting
}

// Issue one 16-row chunk (8 KB) as 16 per-lane async b128 copies.
__device__ __forceinline__ void issue_chunk(const float* __restrict__ hbase,
                                            float* __restrict__ buf, int lane) {
  asm volatile("" ::: "memory");
  const float* g = hbase + lane * 4;
  float* l = buf + lane * 4;
#pragma unroll
  for (int r = 0; r < CHUNK; ++r)
    async_copy_b128(g + r * D_DIM, l + r * PITCH);
  asm volatile("" ::: "memory");
}

// 16x128 rows (in LDS) dot v (in LDS) via V_WMMA_F32_16X16X4_F32.
// Two half-K accumulator chains for WMMA latency hiding, summed at the end.
__device__ __forceinline__ v8f chunk_dot(const float* __restrict__ buf,
                                         const float* __restrict__ s_v,
                                         int lane) {
  const int half = lane >> 4;
  const int row  = lane & 15;
  const float* ap = buf + row * PITCH + 2 * half;
  const float* bp = s_v + 2 * half;
  v8f acc0 = {};
  v8f acc1 = {};
#pragma unroll 4
  for (int d = 0; d < 64; d += 4) {
    v2f a0 = *(const v2f*)(ap + d);
    v2f b0 = *(const v2f*)(bp + d);
    acc0 = __builtin_amdgcn_wmma_f32_16x16x4_f32(false, a0, false, b0,
                                                 (short)0, acc0, false, false);
    v2f a1 = *(const v2f*)(ap + d + 64);
    v2f b1 = *(const v2f*)(bp + d + 64);
    acc1 = __builtin_amdgcn_wmma_f32_16x16x4_f32(false, a1, false, b1,
                                                 (short)0, acc1, false, false);
  }
  return acc0 + acc1;
}

// ---------------------------------------------------------------------------
// Prep: v[d] = W_w^T u1; w2[d] = W_w^T u2; c = W_b·(u1+u2) + u_b.
// ws layout: [0..127]=v  [128..255]=w2  [256]=c
// ---------------------------------------------------------------------------
__global__ __launch_bounds__(128) void prep_kernel(
    const float* __restrict__ W_w, const float* __restrict__ W_b,
    const float* __restrict__ u_w, const float* __restrict__ u_b,
    float* __restrict__ ws) {
  int d = threadIdx.x;  // 0..127
  float vv = 0.f, ww = 0.f;
  for (int hh = 0; hh < 64; ++hh) {
    float w = W_w[hh * D_DIM + d];
    vv = fmaf(u_w[hh], w, vv);
    ww = fmaf(u_w[64 + hh], w, ww);
  }
  ws[d] = vv;
  ws[D_DIM + d] = ww;
  if (d == 0) {
    float c = u_b[0];
    for (int hh = 0; hh < 64; ++hh) c = fmaf(W_b[hh], u_w[hh] + u_w[64 + hh], c);
    ws[2 * D_DIM] = c;
  }
}

// ---------------------------------------------------------------------------
// Main: persistent wave32 per node stream; async double-buffered 16-row
// chunks; fp32 WMMA dot products; wave-level softmax over K=32.
// ---------------------------------------------------------------------------
__global__ __launch_bounds__(64) void attn_kernel(
    const float* __restrict__ h, const float* __restrict__ x,
    const float* __restrict__ ws, float* __restrict__ out, int N) {
  __shared__ float s_v[D_DIM];
  __shared__ float s_w2[D_DIM];
  __shared__ float s_buf[2][2][CHUNK * PITCH];  // [wave][parity], 8.45KB each

  const int tid  = threadIdx.x;
  const int lane = tid & 31;
  const int wave = tid >> 5;

  for (int i = tid; i < D_DIM; i += 64) {
    s_v[i]  = ws[i];
    s_w2[i] = ws[D_DIM + i];
  }
  const float cconst = ws[2 * D_DIM];
  __syncthreads();

  float* buf0 = s_buf[wave][0];
  float* buf1 = s_buf[wave][1];

  const int stride = gridDim.x * 2;  // waves in grid
  int n = blockIdx.x * 2 + wave;     // wave-uniform -> EXEC all-1s for WMMA

  if (n < N) issue_chunk(h + (size_t)n * K_MB * D_DIM, buf0, lane);

  for (; n < N; n += stride) {
    const float* hp = h + (size_t)n * K_MB * D_DIM;

    // per-node scalar s_n = x[n]·w2 + c (VGPR path, overlaps async traffic)
    float4 xv = *(const float4*)(x + (size_t)n * D_DIM + lane * 4);
    float part = xv.x * s_w2[lane * 4 + 0] + xv.y * s_w2[lane * 4 + 1] +
                 xv.z * s_w2[lane * 4 + 2] + xv.w * s_w2[lane * 4 + 3];
    for (int off = 16; off; off >>= 1) part += __shfl_xor(part, off, 32);
    const float sn = part + cconst;

    // pipeline: issue rows 16-31, compute rows 0-15
    issue_chunk(hp + CHUNK * D_DIM, buf1, lane);
    wait_asynccnt<16>();               // chunk0 landed (in-order completion)
    v8f cA = chunk_dot(buf0, s_v, lane);

    // issue next node's rows 0-15, compute rows 16-31
    const int nn = n + stride;
    if (nn < N) {
      issue_chunk(h + (size_t)nn * K_MB * D_DIM, buf0, lane);
      wait_asynccnt<16>();             // chunk1 landed
    } else {
      wait_asynccnt<0>();
    }
    v8f cB = chunk_dot(buf1, s_v, lane);

    // ---- softmax over K=32 ----
    // C/D layout: lane-half hlf holds rows {8*hlf+p} (cA) / {16+8*hlf+p} (cB)
    const int half = lane >> 4;
    float s[16];
    float m = -INFINITY;
#pragma unroll
    for (int p = 0; p < 8; ++p) {
      float t0 = cA[p] + sn;
      t0 = t0 > 0.f ? t0 : 0.01f * t0;  // leaky_relu(0.01)
      float t1 = cB[p] + sn;
      t1 = t1 > 0.f ? t1 : 0.01f * t1;
      s[p] = t0;
      s[8 + p] = t1;
      m = fmaxf(m, fmaxf(t0, t1));
    }
    m = fmaxf(m, __shfl_xor(m, 16, 32));
    float sum = 0.f;
#pragma unroll
    for (int p = 0; p < 16; ++p) {
      s[p] = __expf(s[p] - m);
      sum += s[p];
    }
    sum += __shfl_xor(sum, 16, 32);
    const float inv = 1.0f / sum;

    // each lane stores one distinct row (coalesced 128B segment):
    //  half0: lanes 0-7 -> rows 0-7,  lanes 8-15  -> rows 16-23
    //  half1: lanes16-23-> rows 8-15, lanes 24-31 -> rows 24-31
    const int q   = lane & 7;
    const int hi8 = (lane >> 3) & 1;
    const float wout = (hi8 ? s[8 + q] : s[q]) * inv;
    const int rrow = half ? (hi8 ? 24 + q : 8 + q) : (hi8 ? 16 + q : q);
    out[(size_t)n * K_MB + rrow] = wout;
  }
}

// ---------------------------------------------------------------------------
extern "C" void kernel_launch(void* const* d_in, const int* in_sizes, int n_in,
                              void* d_out, int out_size, void* d_ws,
                              size_t ws_size, hipStream_t stream) {
  const float* h   = (const float*)d_in[0];  // [N, 32, 128]
  const float* x   = (const float*)d_in[1];  // [N, 128]
  const float* W_w = (const float*)d_in[2];  // [64, 128]
  const float* W_b = (const float*)d_in[3];  // [64]
  const float* u_w = (const float*)d_in[4];  // [1, 128]
  const float* u_b = (const float*)d_in[5];  // [1]
  float* out = (float*)d_out;                // [N, 32, 1]
  float* ws  = (float*)d_ws;                 // needs 257 floats

  const int N = in_sizes[1] / D_DIM;

  prep_kernel<<<1, 128, 0, stream>>>(W_w, W_b, u_w, u_b, ws);

  int blocks = (N + 1) / 2;              // 2 waves per 64-thread block
  if (blocks > 4096) blocks = 4096;      // persistent grid-stride waves
  attn_kernel<<<blocks, 64, 0, stream>>>(h, x, ws, out, N);
}